// Discriminator_82068235092697
// MI455X (gfx1250) — compile-verified
//
#include <hip/hip_runtime.h>

#define N_NODES 50000
#define N_EDGES 800000
#define IN_DIM 64
#define HID 128
#define N_GRAPHS 1024
#define NUM_LAYERS 4

typedef __attribute__((ext_vector_type(2))) float v2f;
typedef __attribute__((ext_vector_type(8))) float v8f;

// ---------------------------------------------------------------------------
// Elementwise helpers
// ---------------------------------------------------------------------------
__global__ void fill_zero_kernel(float* __restrict__ p, long n) {
    long i = (long)blockIdx.x * blockDim.x + threadIdx.x;
    if (i < n) p[i] = 0.0f;
}

__global__ void copy_f4_kernel(const float4* __restrict__ s, float4* __restrict__ d, long n4) {
    long i = (long)blockIdx.x * blockDim.x + threadIdx.x;
    if (i < n4) d[i] = s[i];
}

// ---------------------------------------------------------------------------
// Edge scatter-add:  agg[dst[e], :] += h[src[e], :]
// One thread per (edge, 4-float chunk). D = 64 (shift=4) or 128 (shift=5).
// ---------------------------------------------------------------------------
__global__ void scatter_add_edges_kernel(const float* __restrict__ h,
                                         const long long* __restrict__ src,
                                         const long long* __restrict__ dst,
                                         float* __restrict__ agg,
                                         int shift, int D) {
    long t = (long)blockIdx.x * blockDim.x + threadIdx.x;
    long e = t >> shift;
    if (e >= (long)N_EDGES) return;
    int c = ((int)t & ((1 << shift) - 1)) << 2;
    int s = (int)src[e];
    int d = (int)dst[e];
    float4 v = *(const float4*)(h + (long)s * D + c);
    float* p = agg + (long)d * D + c;
    atomicAdd(p + 0, v.x);
    atomicAdd(p + 1, v.y);
    atomicAdd(p + 2, v.z);
    atomicAdd(p + 3, v.w);
}

// ---------------------------------------------------------------------------
// WMMA GEMM: out[nrows x 128] = act( A[nrows x K] @ W[K x 128] + bias + res )
// act: 0=none, 1=relu, 2=leaky(0.01). res may be null; out may alias res,
// and (when K==128) out may alias A (each wave reads only the rows it writes,
// and all reads complete before its epilogue stores).
//
// Block = 256 threads (8 waves); wave computes one 16x128 tile via
// V_WMMA_F32_16X16X4_F32. W is staged into dynamic LDS PRE-SWIZZLED so the
// K-pair each lane needs is contiguous:
//     lds[(k>>1)*256 + col*2 + (k&1)] = W[k][col]
// -> every B fragment is one aligned ds_load_b64 (no register shuffles),
//    and all 8 n-tile fragments share one LDS address with imm offsets.
// ---------------------------------------------------------------------------
__global__ void gemm_wmma_f32_kernel(const float* __restrict__ A,
                                     const float* __restrict__ W,
                                     const float* __restrict__ bias,
                                     const float* __restrict__ res,
                                     float* __restrict__ out,
                                     int nrows, int K, int act) {
    extern __shared__ float ldsW[]; // swizzled [K/2][128][2]
    const int tid = threadIdx.x;
    for (int i = tid; i < K * HID; i += 256) {
        int k = i >> 7;
        int col = i & (HID - 1);
        ldsW[((k >> 1) << 8) + (col << 1) + (k & 1)] = W[i];
    }
    __syncthreads();

    const int lane = tid & 31;
    const int wave = tid >> 5;
    const int row_base = blockIdx.x * 128 + wave * 16;
    const int m = lane & 15;            // row-in-tile for A, col-in-tile for B/C
    const int khalf = (lane >> 4) << 1; // lanes 0-15 -> K+0/1, lanes 16-31 -> K+2/3
    const int arow = row_base + m;
    const bool rok = arow < nrows;
    const long arow_c = rok ? (long)arow : 0;
    const float amask = rok ? 1.0f : 0.0f;

    v8f acc[8] = {}; // 8 N-tiles of 16 cols -> 128 cols

    for (int k = 0; k < K; k += 4) {
        v2f av = *(const v2f*)(A + arow_c * K + k + khalf); // 8B aligned: k+khalf even
        v2f a;
        a.x = av.x * amask;
        a.y = av.y * amask;
        // (k + khalf) is even -> swizzled pair is contiguous, 8B aligned
        const float* brow = ldsW + (((k + khalf) >> 1) << 8) + (m << 1);
#pragma unroll
        for (int nt = 0; nt < 8; ++nt) {
            v2f b = *(const v2f*)(brow + (nt << 5)); // one ds_load_b64, imm offset
            acc[nt] = __builtin_amdgcn_wmma_f32_16x16x4_f32(
                false, a, false, b, (short)0, acc[nt], false, false);
        }
    }

    const int rhalf = (lane >> 4) << 3; // lanes 16-31 hold rows r+8
#pragma unroll
    for (int nt = 0; nt < 8; ++nt) {
        const int col = nt * 16 + m;
        const float bv = bias ? bias[col] : 0.0f;
#pragma unroll
        for (int r = 0; r < 8; ++r) {
            const int orow = row_base + r + rhalf;
            if (orow < nrows) {
                float v = acc[nt][r] + bv;
                if (res) v += res[(long)orow * HID + col];
                if (act == 1) v = v > 0.0f ? v : 0.0f;
                else if (act == 2) v = v > 0.0f ? v : 0.01f * v;
                out[(long)orow * HID + col] = v;
            }
        }
    }
}

// ---------------------------------------------------------------------------
// Mean-pool pieces
// ---------------------------------------------------------------------------
__global__ void pool_sum_kernel(const float* __restrict__ h,
                                const long long* __restrict__ batch,
                                float* __restrict__ sums,
                                float* __restrict__ counts) {
    long t = (long)blockIdx.x * blockDim.x + threadIdx.x;
    long node = t >> 5; // 32 chunks of 4 floats = 128 dims
    if (node >= (long)N_NODES) return;
    int c = ((int)t & 31) << 2;
    int g = (int)batch[node];
    float4 v = *(const float4*)(h + node * HID + c);
    float* p = sums + (long)g * HID + c;
    atomicAdd(p + 0, v.x);
    atomicAdd(p + 1, v.y);
    atomicAdd(p + 2, v.z);
    atomicAdd(p + 3, v.w);
    if (c == 0) atomicAdd(&counts[g], 1.0f);
}

__global__ void pool_div_kernel(float* __restrict__ sums, const float* __restrict__ counts) {
    int t = blockIdx.x * blockDim.x + threadIdx.x;
    if (t >= N_GRAPHS * HID) return;
    int g = t >> 7;
    sums[t] = sums[t] / fmaxf(counts[g], 1.0f);
}

// Final 128 -> 1 matvec: one wave per graph.
__global__ void head_final_kernel(const float* __restrict__ hT,
                                  const float* __restrict__ fW2,
                                  const float* __restrict__ fb2,
                                  float* __restrict__ out) {
    int g = blockIdx.x * (blockDim.x >> 5) + (threadIdx.x >> 5);
    int lane = threadIdx.x & 31;
    if (g >= N_GRAPHS) return;
    float s = 0.0f;
#pragma unroll
    for (int d = lane; d < HID; d += 32) s += hT[(long)g * HID + d] * fW2[d];
#pragma unroll
    for (int off = 16; off > 0; off >>= 1) s += __shfl_xor(s, off, 32);
    if (lane == 0) out[g] = s + fb2[0];
}

// ---------------------------------------------------------------------------
// Orchestration
// ---------------------------------------------------------------------------
extern "C" void kernel_launch(void* const* d_in, const int* in_sizes, int n_in,
                              void* d_out, int out_size, void* d_ws, size_t ws_size,
                              hipStream_t stream) {
    (void)in_sizes; (void)n_in; (void)out_size; (void)ws_size;

    const float*     x      = (const float*)d_in[0];
    const long long* ei     = (const long long*)d_in[1];
    const long long* batch  = (const long long*)d_in[2];
    const float*     l0_W1  = (const float*)d_in[3];
    const float*     l0_b1  = (const float*)d_in[4];
    const float*     l0_W2  = (const float*)d_in[5];
    const float*     l0_b2  = (const float*)d_in[6];
    const float*     l0_Wres= (const float*)d_in[7];
    const float*     Ws1    = (const float*)d_in[8];
    const float*     bs1    = (const float*)d_in[9];
    const float*     Ws2    = (const float*)d_in[10];
    const float*     bs2    = (const float*)d_in[11];
    const float*     fW1    = (const float*)d_in[12];
    const float*     fb1    = (const float*)d_in[13];
    const float*     fW2    = (const float*)d_in[14];
    const float*     fb2    = (const float*)d_in[15];
    float* out = (float*)d_out;

    const long long* src = ei;
    const long long* dst = ei + N_EDGES;

    // Workspace carve (floats): AGG, HA, HB (N x 128 each), SUMS, CNT
    float* ws = (float*)d_ws;
    size_t off = 0;
    float* AGG = ws + off; off += (size_t)N_NODES * HID;
    float* HA  = ws + off; off += (size_t)N_NODES * HID;
    float* HB  = ws + off; off += (size_t)N_NODES * HID;
    float* SUMS = ws + off; off += (size_t)N_GRAPHS * HID;
    float* CNT  = ws + off; off += (size_t)N_GRAPHS;

    const int BLK = 256;
    const int gemm_blocks_nodes = (N_NODES + 127) / 128;
    const size_t lds64  = (size_t)IN_DIM * HID * sizeof(float); // 32 KB
    const size_t lds128 = (size_t)HID * HID * sizeof(float);    // 64 KB

    // ---------------- Layer 0 (in=64 -> 128, linear residual) ----------------
    {
        long n4 = (long)N_NODES * IN_DIM / 4;
        copy_f4_kernel<<<(int)((n4 + BLK - 1) / BLK), BLK, 0, stream>>>(
            (const float4*)x, (float4*)AGG, n4);
        long sthreads = (long)N_EDGES << 4; // D=64 -> 16 float4 chunks
        scatter_add_edges_kernel<<<(int)((sthreads + BLK - 1) / BLK), BLK, 0, stream>>>(
            x, src, dst, AGG, 4, IN_DIM);
        // T = relu((x+agg) @ W1 + b1) -> HB
        gemm_wmma_f32_kernel<<<gemm_blocks_nodes, BLK, lds64, stream>>>(
            AGG, l0_W1, l0_b1, nullptr, HB, N_NODES, IN_DIM, 1);
        // RES = x @ Wres -> HA
        gemm_wmma_f32_kernel<<<gemm_blocks_nodes, BLK, lds64, stream>>>(
            x, l0_Wres, nullptr, nullptr, HA, N_NODES, IN_DIM, 0);
        // H = relu(T @ W2 + b2 + RES) -> HA (out aliases res: per-thread read-then-write)
        gemm_wmma_f32_kernel<<<gemm_blocks_nodes, BLK, lds128, stream>>>(
            HB, l0_W2, l0_b2, HA, HA, N_NODES, HID, 1);
    }

    float* Hcur = HA;
    float* Hoth = HB;

    // ---------------- Layers 1..3 (128 -> 128, identity residual) ------------
    for (int i = 0; i < NUM_LAYERS - 1; ++i) {
        long n4 = (long)N_NODES * HID / 4;
        copy_f4_kernel<<<(int)((n4 + BLK - 1) / BLK), BLK, 0, stream>>>(
            (const float4*)Hcur, (float4*)AGG, n4);
        long sthreads = (long)N_EDGES << 5; // D=128 -> 32 float4 chunks
        scatter_add_edges_kernel<<<(int)((sthreads + BLK - 1) / BLK), BLK, 0, stream>>>(
            Hcur, src, dst, AGG, 5, HID);
        // T = relu((h+agg) @ Ws1[i] + bs1[i]) -> AGG (safe alias, K==128)
        gemm_wmma_f32_kernel<<<gemm_blocks_nodes, BLK, lds128, stream>>>(
            AGG, Ws1 + (size_t)i * HID * HID, bs1 + (size_t)i * HID, nullptr,
            AGG, N_NODES, HID, 1);
        // Hnext = relu(T @ Ws2[i] + bs2[i] + Hcur) -> Hoth
        gemm_wmma_f32_kernel<<<gemm_blocks_nodes, BLK, lds128, stream>>>(
            AGG, Ws2 + (size_t)i * HID * HID, bs2 + (size_t)i * HID, Hcur,
            Hoth, N_NODES, HID, 1);
        float* tmp = Hcur; Hcur = Hoth; Hoth = tmp;
    }

    // ---------------- Global mean pool ----------------
    {
        long nz = (long)N_GRAPHS * HID + N_GRAPHS; // SUMS + CNT are contiguous
        fill_zero_kernel<<<(int)((nz + BLK - 1) / BLK), BLK, 0, stream>>>(SUMS, nz);
        long pthreads = (long)N_NODES * 32;
        pool_sum_kernel<<<(int)((pthreads + BLK - 1) / BLK), BLK, 0, stream>>>(
            Hcur, batch, SUMS, CNT);
        pool_div_kernel<<<(N_GRAPHS * HID + BLK - 1) / BLK, BLK, 0, stream>>>(SUMS, CNT);
    }

    // ---------------- Head: leaky(pooled @ fW1 + fb1) @ fW2 + fb2 ------------
    {
        const int gemm_blocks_g = (N_GRAPHS + 127) / 128; // 8
        // headT -> AGG (free now)
        gemm_wmma_f32_kernel<<<gemm_blocks_g, BLK, lds128, stream>>>(
            SUMS, fW1, fb1, nullptr, AGG, N_GRAPHS, HID, 2);
        head_final_kernel<<<(N_GRAPHS * 32) / BLK, BLK, 0, stream>>>(AGG, fW2, fb2, out);
    }
}